// DMDMachine_17849884082186
// MI455X (gfx1250) — compile-verified
//
#include <hip/hip_runtime.h>

// Reference shapes: B=512, N=256, D=64, T=128
#define BB 512
#define NN 256
#define DD 64
#define TT 128

typedef float v2f __attribute__((ext_vector_type(2)));
typedef float v8f __attribute__((ext_vector_type(8)));
typedef unsigned int v4u __attribute__((ext_vector_type(4)));
typedef int v4i __attribute__((ext_vector_type(4)));
typedef int v8i __attribute__((ext_vector_type(8)));

#if defined(__has_builtin)
# if __has_builtin(__builtin_amdgcn_wmma_f32_16x16x4_f32)
#  define HAS_WMMA_F32X4 1
# endif
# if __has_builtin(__builtin_amdgcn_tensor_load_to_lds)
#  define HAS_TDM 1
# endif
#endif

// D(16x16,f32) = A(16x4,f32) * B(4x16,f32) + C  --- V_WMMA_F32_16X16X4_F32
__device__ __forceinline__ v8f wmma4(v2f a, v2f b, v8f c) {
#if defined(HAS_WMMA_F32X4)
  return __builtin_amdgcn_wmma_f32_16x16x4_f32(false, a, false, b, (short)0, c,
                                               false, false);
#else
  c[0] += a.x * b.x + a.y * b.y;   // placeholder (histogram would show wmma=0)
  return c;
#endif
}

// TDM: load a 2D row-major tile (f32) from global to LDS.
//   D# group0: [1:0]=count=1, [63:32]=lds_addr, [120:64]=global_addr, [127:126]=type=2
//   D# group1: [17:16]=data_size(2 -> 4B), [79:48]=tensor_dim0, [111:80]=tensor_dim1,
//              [127:112]=tile_dim0, [143:128]=tile_dim1, [207:160]=dim0_stride
__device__ __forceinline__ void tdm_load_2d(unsigned lds_off, const void* gaddr,
                                            unsigned dim0, unsigned dim1,
                                            unsigned tile0, unsigned tile1,
                                            unsigned stride0) {
#if defined(HAS_TDM)
  const unsigned long long ga = (unsigned long long)(uintptr_t)gaddr;
  v4u g0;
  g0.x = 1u;                                                  // count=1
  g0.y = lds_off;                                             // lds_addr (bytes)
  g0.z = (unsigned)ga;                                        // global_addr[31:0]
  g0.w = (unsigned)((ga >> 32) & 0x01FFFFFFu) | (2u << 30);   // [56:32] | type=2
  v8i g1;
  g1[0] = (int)(2u << 16);                                    // data_size = 4B
  g1[1] = (int)((dim0 & 0xFFFFu) << 16);                      // tensor_dim0 lo
  g1[2] = (int)(((dim0 >> 16) & 0xFFFFu) | ((dim1 & 0xFFFFu) << 16));
  g1[3] = (int)(((dim1 >> 16) & 0xFFFFu) | ((tile0 & 0xFFFFu) << 16));
  g1[4] = (int)(tile1 & 0xFFFFu);                             // tile_dim1, dim2=0
  g1[5] = (int)stride0;                                       // dim0_stride[31:0]
  g1[6] = 0;
  g1[7] = 0;
  const v4i z4 = {0, 0, 0, 0};
#if (__clang_major__ >= 23)
  const v8i z8 = {0, 0, 0, 0, 0, 0, 0, 0};
  __builtin_amdgcn_tensor_load_to_lds(g0, g1, z4, z4, z8, 0);
#else
  __builtin_amdgcn_tensor_load_to_lds(g0, g1, z4, z4, 0);
#endif
#else
  (void)lds_off; (void)gaddr; (void)dim0; (void)dim1;
  (void)tile0; (void)tile1; (void)stride0;
#endif
}

__global__ void dmd_zero_scalars(float* dmd, float* ae, float* pred) {
  if (threadIdx.x == 0) { *dmd = 0.f; *ae = 0.f; *pred = 0.f; }
}

// One workgroup (256 threads = 8 waves) per batch. Everything lives in LDS.
//
// LDS map (dynamic, 288 KB):
//   sWe [0,64K)  sWd [64K,128K)  sY [128K,160K)  sX [160K,288K)
//   after phase 2/3 retire sWe/sWd, alias:
//   sG [0,16640) sC [16640,33280) sA [33280,49920) sV [49920,50432)
//   sRed [50432,50464)  sPred [64K, 96K)
__global__ void __launch_bounds__(256)
dmd_mega(const float* __restrict__ x, const float* __restrict__ We,
         const float* __restrict__ Wd, float* __restrict__ x_ae,
         float* __restrict__ y_out, float* __restrict__ y_pred,
         float* __restrict__ dmd_loss, float* __restrict__ ae_loss,
         float* __restrict__ pred_loss)
{
  extern __shared__ char smem[];
  float* sWe   = (float*)(smem);            // 64 x 256, stride 256
  float* sWd   = (float*)(smem + 65536);    // 256 x 64, stride 64
  float* sY    = (float*)(smem + 131072);   // 64 x 128, stride 128
  float* sX    = (float*)(smem + 163840);   // 256 x 128, stride 128
  float* sG    = (float*)(smem);            // 64 x 64, stride 65 (lower = L)
  float* sC    = (float*)(smem + 16640);    // 64 x 64, stride 65 (-> Z)
  float* sA    = (float*)(smem + 33280);    // 64 x 64, stride 65 (DMD operator)
  float* sV    = (float*)(smem + 49920);    // 2 x 64 ping-pong vectors
  float* sRed  = (float*)(smem + 50432);    // reductions
  float* sPred = (float*)(smem + 65536);    // 64 x 128 y_pred staging

  const int tid  = threadIdx.x;
  const int w    = tid >> 5;     // wave 0..7
  const int ln   = tid & 31;
  const int half = ln >> 4;      // ISA WMMA lane-group
  const int l    = ln & 15;
  const int nb   = w * 16;       // this wave's N(=t) tile base
  const int b    = blockIdx.x;
  const float* xb = x + (size_t)b * (NN * TT);

  // ---------- phase 1: stage We, Wd, x_b into LDS via Tensor Data Mover ----
#if defined(HAS_TDM)
  if (tid < 32) {                 // wave 0 drives the TDM
    tdm_load_2d(0u,       We, NN, DD, NN, DD, NN);   // We: 64x256
    tdm_load_2d(65536u,   Wd, DD, NN, DD, NN, DD);   // Wd: 256x64
    tdm_load_2d(163840u,  xb, TT, NN, TT, NN, TT);   // x_b: 256x128
    __builtin_amdgcn_s_wait_tensorcnt(0);
  }
#else
  {
    const float4* We4 = (const float4*)We;
    const float4* Wd4 = (const float4*)Wd;
    const float4* xb4 = (const float4*)xb;
    float4* sWe4 = (float4*)sWe;
    float4* sWd4 = (float4*)sWd;
    float4* sX4  = (float4*)sX;
    for (int i = tid; i < (DD * NN) / 4; i += 256) sWe4[i] = We4[i];
    for (int i = tid; i < (NN * DD) / 4; i += 256) sWd4[i] = Wd4[i];
    for (int i = tid; i < (NN * TT) / 4; i += 256) sX4[i]  = xb4[i];
  }
#endif
  __syncthreads();

  // ---------- phase 2: y = We(64x256) @ x_b(256x128), WMMA ------------------
  // software-pipelined: ds loads for step k+4 issue before wmmas of step k
  {
    v8f acc[4] = {};
    v2f bf, af[4];
    {
      const int kk = 2 * half;
      bf.x = sX[(kk    ) * TT + nb + l];
      bf.y = sX[(kk + 1) * TT + nb + l];
      #pragma unroll
      for (int mt = 0; mt < 4; ++mt) {
        af[mt].x = sWe[(mt * 16 + l) * NN + kk];
        af[mt].y = sWe[(mt * 16 + l) * NN + kk + 1];
      }
    }
    for (int k = 0; k < NN; k += 4) {
      const int kn = (k + 4 < NN) ? (k + 4) : 0;   // dummy reload on last iter
      const int kk = kn + 2 * half;
      v2f bf_n, af_n[4];
      bf_n.x = sX[(kk    ) * TT + nb + l];
      bf_n.y = sX[(kk + 1) * TT + nb + l];
      #pragma unroll
      for (int mt = 0; mt < 4; ++mt) {
        af_n[mt].x = sWe[(mt * 16 + l) * NN + kk];
        af_n[mt].y = sWe[(mt * 16 + l) * NN + kk + 1];
      }
      #pragma unroll
      for (int mt = 0; mt < 4; ++mt)
        acc[mt] = wmma4(af[mt], bf, acc[mt]);
      bf = bf_n;
      #pragma unroll
      for (int mt = 0; mt < 4; ++mt) af[mt] = af_n[mt];
    }
    #pragma unroll
    for (int mt = 0; mt < 4; ++mt) {
      #pragma unroll
      for (int r = 0; r < 8; ++r) {
        const int m = mt * 16 + r + 8 * half;
        sY[m * TT + nb + l] = acc[mt][r];
      }
    }
  }
  __syncthreads();

  // coalesced b128 store of y to global
  {
    const float4* sY4 = (const float4*)sY;
    float4* yo4 = (float4*)(y_out + (size_t)b * DD * TT);
    for (int i = tid; i < (DD * TT) / 4; i += 256) yo4[i] = sY4[i];
  }

  // ---------- phase 3: x_ae = Wd(256x64) @ y(64x128) + ae_loss (t==0 col) ---
  {
    float aeAcc = 0.f;
    for (int mt = 0; mt < 16; ++mt) {
      const int mb = mt * 16;
      v8f acc = {};
      #pragma unroll
      for (int k = 0; k < DD; k += 4) {
        const int kk = k + 2 * half;
        v2f a, bf;
        a.x  = sWd[(mb + l) * DD + kk];
        a.y  = sWd[(mb + l) * DD + kk + 1];
        bf.x = sY[(kk    ) * TT + nb + l];
        bf.y = sY[(kk + 1) * TT + nb + l];
        acc = wmma4(a, bf, acc);
      }
      #pragma unroll
      for (int r = 0; r < 8; ++r) {
        const int m = mb + r + 8 * half;
        const float v = acc[r];
        x_ae[((size_t)b * NN + m) * TT + nb + l] = v;
        if (nb + l == 0) { const float d0 = sX[m * TT] - v; aeAcc += d0 * d0; }
      }
    }
    if (nb + l == 0)
      atomicAdd(ae_loss, aeAcc * (1.0f / ((float)BB * (float)NN)));
  }
  __syncthreads();   // retire sWe/sWd regions

  if (tid < 8) sRed[tid] = 0.f;

  // ---------- phase 4: G = Ym*Ym^T, C = Yp*Ym^T (64x64, K=127), WMMA --------
  // software-pipelined like phase 2; K padded 127->128 with masked loads
  {
    #pragma unroll
    for (int tt2 = 0; tt2 < 2; ++tt2) {
      const int t4 = 2 * w + tt2;          // 0..15 over 4x4 tile grid
      const int mb = (t4 >> 2) * 16;
      const int cb = (t4 & 3) * 16;
      v8f gAcc = {}, cAcc = {};
      v2f bf, ag, ac;
      {
        const int kk = 2 * half;
        bf.x = sY[(cb + l) * TT + kk    ];
        bf.y = sY[(cb + l) * TT + kk + 1];
        ag.x = sY[(mb + l) * TT + kk    ];
        ag.y = sY[(mb + l) * TT + kk + 1];
        ac.x = sY[(mb + l) * TT + kk + 1];
        ac.y = sY[(mb + l) * TT + kk + 2];
      }
      for (int k = 0; k < TT; k += 4) {
        const int kn = (k + 4 < TT) ? (k + 4) : 0;
        const int kk = kn + 2 * half;
        const bool v0 = kk < (TT - 1);
        const bool v1 = (kk + 1) < (TT - 1);
        v2f bfn, agn, acn;
        bfn.x = v0 ? sY[(cb + l) * TT + kk    ] : 0.f;   // B[k][n] = Ym[n][k]
        bfn.y = v1 ? sY[(cb + l) * TT + kk + 1] : 0.f;
        agn.x = v0 ? sY[(mb + l) * TT + kk    ] : 0.f;   // A_G[m][k] = Ym[m][k]
        agn.y = v1 ? sY[(mb + l) * TT + kk + 1] : 0.f;
        acn.x = v0 ? sY[(mb + l) * TT + kk + 1] : 0.f;   // A_C[m][k] = Yp[m][k]
        acn.y = v1 ? sY[(mb + l) * TT + kk + 2] : 0.f;
        gAcc = wmma4(ag, bf, gAcc);
        cAcc = wmma4(ac, bf, cAcc);
        bf = bfn; ag = agn; ac = acn;
      }
      #pragma unroll
      for (int r = 0; r < 8; ++r) {
        const int m = mb + r + 8 * half;
        const int n = cb + l;
        sG[m * 65 + n] = gAcc[r] + (m == n ? 1e-6f : 0.f);  // tiny ridge
        sC[m * 65 + n] = cAcc[r];
      }
    }
  }
  __syncthreads();

  // ---------- sum(Yp^2) reduction (ds float atomics) ----------
  {
    float s = 0.f;
    for (int i = tid; i < DD * (TT - 1); i += 256) {
      const int d = i / (TT - 1), t = i % (TT - 1);
      const float v = sY[d * TT + t + 1];
      s += v * v;
    }
    atomicAdd(&sRed[0], s);
  }

  // ---------- phase 5: in-place Cholesky G = L L^T (lower, stride 65) ------
  for (int k = 0; k < DD; ++k) {
    __syncthreads();
    if (tid == 0) sG[k * 65 + k] = sqrtf(sG[k * 65 + k]);
    __syncthreads();
    const float inv = 1.0f / sG[k * 65 + k];
    for (int i = k + 1 + tid; i < DD; i += 256) sG[i * 65 + k] *= inv;
    __syncthreads();
    const int rem = DD - 1 - k;
    for (int idx = tid; idx < rem * rem; idx += 256) {
      const int i = k + 1 + idx / rem;
      const int j = k + 1 + idx % rem;
      if (j <= i) sG[i * 65 + j] -= sG[i * 65 + k] * sG[j * 65 + k];
    }
  }
  __syncthreads();

  // ---------- phase 6: A = C G^{-1}  (row i: L z = c_i ; L^T a_i = z) ------
  if (tid < DD) {
    const int i = tid;
    for (int j = 0; j < DD; ++j) {                 // forward: Z overwrites C
      float s = sC[i * 65 + j];
      for (int p = 0; p < j; ++p) s -= sG[j * 65 + p] * sC[i * 65 + p];
      sC[i * 65 + j] = s / sG[j * 65 + j];
    }
    for (int j = DD - 1; j >= 0; --j) {            // backward -> A
      float s = sC[i * 65 + j];
      for (int p = j + 1; p < DD; ++p) s -= sG[p * 65 + j] * sA[i * 65 + p];
      sA[i * 65 + j] = s / sG[j * 65 + j];
    }
  }
  __syncthreads();

  // ---------- dmd_loss contribution: sum(Yp^2) - ||Z||_F^2 ----------
  {
    float s = 0.f;
    for (int i = tid; i < DD * DD; i += 256) {
      const float v = sC[(i >> 6) * 65 + (i & 63)];
      s += v * v;
    }
    atomicAdd(&sRed[1], s);
  }
  __syncthreads();
  if (tid == 0)
    atomicAdd(dmd_loss, (sRed[0] - sRed[1]) * (1.0f / (float)BB));

  // ---------- phase 7: y_pred rollout (into LDS) + pred_loss ----------
  if (tid < DD) {
    const float v0 = sY[tid * TT];
    sV[tid] = v0;
    sPred[tid * TT] = v0;
  }
  __syncthreads();
  {
    float pAcc = 0.f;
    for (int t = 1; t < TT; ++t) {
      const float* prev = sV + (((t & 1) ^ 1) * DD);
      float*       cur  = sV + ((t & 1) * DD);
      if (tid < DD) {
        float s = 0.f;
        #pragma unroll 8
        for (int k = 0; k < DD; ++k) s += sA[tid * 65 + k] * prev[k];
        cur[tid] = s;
        sPred[tid * TT + t] = s;
        const float d = s - sY[tid * TT + t];
        pAcc += d * d;
      }
      __syncthreads();
    }
    if (tid < DD) atomicAdd(&sRed[2], pAcc);
  }
  __syncthreads();
  if (tid == 0)
    atomicAdd(pred_loss, sRed[2] * (1.0f / ((float)BB * (float)DD * (float)TT)));

  // coalesced b128 store of y_pred to global
  {
    const float4* sP4 = (const float4*)sPred;
    float4* yp4 = (float4*)(y_pred + (size_t)b * DD * TT);
    for (int i = tid; i < (DD * TT) / 4; i += 256) yp4[i] = sP4[i];
  }
}

extern "C" void kernel_launch(void* const* d_in, const int* in_sizes, int n_in,
                              void* d_out, int out_size, void* d_ws, size_t ws_size,
                              hipStream_t stream) {
  (void)in_sizes; (void)n_in; (void)out_size; (void)d_ws; (void)ws_size;
  const float* x  = (const float*)d_in[0];
  const float* We = (const float*)d_in[1];
  const float* Wd = (const float*)d_in[2];

  float* o      = (float*)d_out;
  float* x_ae   = o;                                   // 512*256*128
  float* y      = x_ae + (size_t)BB * NN * TT;         // 512*64*128
  float* dmd    = y    + (size_t)BB * DD * TT;         // scalar
  float* ae     = dmd + 1;                             // scalar
  float* ypred  = ae + 1;                              // 512*64*128
  float* pred   = ypred + (size_t)BB * DD * TT;        // scalar

  dmd_zero_scalars<<<1, 32, 0, stream>>>(dmd, ae, pred);

  const size_t lds_bytes = 294912;                     // 288 KB <= 320 KB/WGP
  dmd_mega<<<BB, 256, lds_bytes, stream>>>(x, We, Wd, x_ae, y, ypred,
                                           dmd, ae, pred);
}